// PWM_163208757262
// MI455X (gfx1250) — compile-verified
//
#include <hip/hip_runtime.h>

typedef __attribute__((ext_vector_type(2))) float v2f;
typedef __attribute__((ext_vector_type(8))) float v8f;

#define NPWM    512
#define FS      19
#define LEN     1000
#define OUTPOS  982      // LEN - FS + 1
#define PTILES  62       // ceil(982 / 16)
#define KSTEPS  19       // K = 4*19 = 76, 4 per WMMA
#define NEG_INF (-3.402823466e38f)

static __device__ __forceinline__ v8f wmma_f32_16x16x4(v2f a, v2f b, v8f c) {
  // D(16x16,f32) = A(16x4,f32) x B(4x16,f32) + C  — CDNA5 V_WMMA_F32_16X16X4_F32
  return __builtin_amdgcn_wmma_f32_16x16x4_f32(
      /*neg_a=*/false, a, /*neg_b=*/false, b,
      /*c_mod=*/(short)0, c, /*reuse_a=*/false, /*reuse_b=*/false);
}

__global__ __launch_bounds__(256) void pwm_scan_max_kernel(
    const float* __restrict__ x,      // [128][4][1000]
    const float* __restrict__ pwms,   // [512][4][19]
    float* __restrict__ out) {        // [128][512]
  __shared__ float xT[4096];          // position-major x[b]: xT[pos*4 + c], pos padded to 1024
  __shared__ float red[8][16];

  const int b     = blockIdx.y;
  const int f0    = blockIdx.x * 16;      // 16 filters per block
  const int tid   = threadIdx.x;
  const int lane  = tid & 31;
  const int wave  = tid >> 5;
  const int n     = lane & 15;            // filter-in-tile (B/D column)
  const int cbase = (lane < 16) ? 0 : 2;  // K-pair owned by this half-wave

  // ---- stage x[b] into LDS, transposed to position-major, zero-padded ----
  const float* xb = x + (size_t)b * 4 * LEN;
  for (int idx = tid; idx < 4096; idx += 256) {
    int pos = idx >> 2, c = idx & 3;
    xT[idx] = (pos < LEN) ? xb[c * LEN + pos] : 0.0f;
  }

  // ---- load this lane's B-matrix fragments: forward + reverse-complement ----
  // kk = k*4 + c ; forward: W[f, c, k] ; revcomp: W[f, 3-c, 18-k]
  const float* wrow = pwms + (size_t)(f0 + n) * (4 * FS);
  v2f bF[KSTEPS], bR[KSTEPS];
#pragma unroll
  for (int g = 0; g < KSTEPS; ++g) {
    bF[g].x = wrow[cbase * FS + g];
    bF[g].y = wrow[(cbase + 1) * FS + g];
    bR[g].x = wrow[(3 - cbase) * FS + (FS - 1 - g)];
    bR[g].y = wrow[(2 - cbase) * FS + (FS - 1 - g)];
  }

  __syncthreads();

  float rmax = NEG_INF;
  // A-layout: lane holds row M = lane&15; ds_load_b64 per K-step (8B aligned)
  const float* ap_base = &xT[0];
  for (int pt = wave; pt < PTILES; pt += 8) {
    const int pos0 = pt * 16;
    v8f accF = {};
    v8f accR = {};
    const float* ap = ap_base + (pos0 + n) * 4 + cbase;
#pragma unroll
    for (int g = 0; g < KSTEPS; ++g) {
      v2f a = *(const v2f*)(ap + g * 4);       // x[pos0+n, k=g, c=cbase..cbase+1]
      accF = wmma_f32_16x16x4(a, bF[g], accF); // forward strand
      accR = wmma_f32_16x16x4(a, bR[g], accR); // reverse-complement strand
    }
    // D-layout: reg r holds position pos0 + r + 8*(lane>=16); mask padded t>=982
    const int mbase = pos0 + ((lane >= 16) ? 8 : 0);
#pragma unroll
    for (int r = 0; r < 8; ++r) {
      float v = fmaxf(accF[r], accR[r]);
      rmax = (mbase + r < OUTPOS) ? fmaxf(rmax, v) : rmax;
    }
  }

  // combine the two half-wave position groups, then reduce across waves
  rmax = fmaxf(rmax, __shfl_xor(rmax, 16, 32));
  if (lane < 16) red[wave][n] = rmax;
  __syncthreads();
  if (tid < 16) {
    float m = red[0][tid];
#pragma unroll
    for (int w = 1; w < 8; ++w) m = fmaxf(m, red[w][tid]);
    out[(size_t)b * NPWM + f0 + tid] = m;
  }
}

extern "C" void kernel_launch(void* const* d_in, const int* in_sizes, int n_in,
                              void* d_out, int out_size, void* d_ws, size_t ws_size,
                              hipStream_t stream) {
  const float* x    = (const float*)d_in[0];  // (128, 4, 1000) f32
  const float* pwms = (const float*)d_in[1];  // (512, 4, 19) f32
  float* out        = (float*)d_out;          // (128, 512) f32

  dim3 grid(NPWM / 16, 128);                  // (filter tiles, batch)
  pwm_scan_max_kernel<<<grid, 256, 0, stream>>>(x, pwms, out);
}